// glp_rotation_pool_3813930959531
// MI455X (gfx1250) — compile-verified
//
#include <hip/hip_runtime.h>
#include <stdint.h>

namespace {

constexpr int H = 64, W = 64, A = 32, G = 8;
constexpr int BCS = H * W * A;          // input  bc-stride (floats) = 131072
constexpr int OCS = H * W * G;          // output bc-stride (floats) = 32768
constexpr int ITER = 32;                // bc values per wave (8 waves/block -> 256 bc)
constexpr int TAPF = 9;                 // padded tap row (floats) -> conflict-free DS reads
constexpr int BUF = 16 * TAPF;          // one staging buffer: 16 taps * 9 floats = 576 B

// theta_j = -11.25 * j degrees, j = angle % 4
__device__ __forceinline__ void rot_cs(int j, float& c, float& s) {
  c = (j == 0) ? 1.0f
      : (j == 1) ? 0.98078528040323044913f
      : (j == 2) ? 0.92387953251128675613f
                 : 0.83146961230254523708f;
  s = (j == 0) ? 0.0f
      : (j == 1) ? -0.19509032201612826785f
      : (j == 2) ? -0.38268343236508977173f
                 : -0.55557023301960222474f;
}

__global__ __launch_bounds__(256) void rotpool(const float* __restrict__ in,
                                               float* __restrict__ out) {
  // per-wave double-buffered staging: [wave][2 buffers * 16 taps * 9 floats]
  __shared__ float tile[8][2 * BUF];

  const int lane = threadIdx.x & 31;
  const int wv   = threadIdx.x >> 5;
  const int yx   = blockIdx.x;                 // 0..4095 : one (y,x) per block
  const int x = yx & (W - 1);
  const int y = yx >> 6;
  const int bc0 = wv * ITER;                   // this wave's bc range [bc0, bc0+32)
  const float xr = (float)x - 31.5f;
  const float yr = (float)y - 31.5f;

  // ---- one-time per-lane setup (amortized over 32 bc iterations) ----
  // Issue-slot geometry: slot sl covers tap = sl*4 + (lane>>3), word m = lane&7,
  // fetching angle a = 4m + j of tap pixel (clamped; zero-fill via weights).
  uint32_t voff[4];   // 32-bit global byte offsets (GVS vaddr), fixed per lane
  uint32_t ldst[4];   // LDS byte addresses for buffer 0, fixed per lane
#pragma unroll
  for (int sl = 0; sl < 4; ++sl) {
    const int tap = sl * 4 + (lane >> 3);
    const int j = tap >> 2, t = tap & 3;
    float c, s; rot_cs(j, c, s);
    const float xs = c * xr + s * yr + 31.5f;
    const float ys = -s * xr + c * yr + 31.5f;
    int xi = (int)floorf(xs) + (t & 1);
    int yi = (int)floorf(ys) + (t >> 1);
    xi = min(max(xi, 0), W - 1);
    yi = min(max(yi, 0), H - 1);
    const int m = lane & 7;
    voff[sl] = (uint32_t)(((yi * W + xi) * A + 4 * m + j) * 4);
    ldst[sl] = (uint32_t)(uintptr_t)&tile[wv][tap * TAPF + m];
  }

  // Per-lane bilinear weights for angle a = lane, rotation j = a&3.
  const int a = lane, j = a & 3, m = a >> 2;
  float c, s; rot_cs(j, c, s);
  const float xs = c * xr + s * yr + 31.5f;
  const float ys = -s * xr + c * yr + 31.5f;
  const float x0f = floorf(xs), y0f = floorf(ys);
  const int x0 = (int)x0f, y0 = (int)y0f;
  const float wx = xs - x0f, wy = ys - y0f;
  const float vx0 = (x0 >= 0 && x0 < W) ? 1.0f : 0.0f;
  const float vx1 = (x0 + 1 >= 0 && x0 + 1 < W) ? 1.0f : 0.0f;
  const float vy0 = (y0 >= 0 && y0 < H) ? 1.0f : 0.0f;
  const float vy1 = (y0 + 1 >= 0 && y0 + 1 < H) ? 1.0f : 0.0f;
  const float w00 = (1.0f - wx) * (1.0f - wy) * vx0 * vy0;
  const float w01 = wx * (1.0f - wy) * vx1 * vy0;
  const float w10 = (1.0f - wx) * wy * vx0 * vy1;
  const float w11 = wx * wy * vx1 * vy1;

  const float* __restrict__ rp = &tile[wv][(j * 4) * TAPF + m];   // buffer-0 read base
  float* __restrict__ outp = out + (size_t)bc0 * OCS + (size_t)yx * G + (a >> 2);

  // ---- async issue: 4 x GLOBAL_LOAD_ASYNC_TO_LDS_B32, GVS form
  //      (uniform SGPR64 base per bc, fixed per-lane 32-bit voffset) ----
  auto issue = [&](int it, uint32_t bofB) {
    const uint64_t sb = (uint64_t)(uintptr_t)(in + (size_t)(bc0 + it) * BCS);
#pragma unroll
    for (int sl = 0; sl < 4; ++sl) {
      asm volatile("global_load_async_to_lds_b32 %0, %1, %2"
                   :
                   : "v"(ldst[sl] + bofB), "v"(voff[sl]), "s"(sb)
                   : "memory");
    }
  };

  auto compute = [&](int it, int bofF) {
    const float v0 = rp[bofF + 0 * TAPF];
    const float v1 = rp[bofF + 1 * TAPF];
    const float v2 = rp[bofF + 2 * TAPF];
    const float v3 = rp[bofF + 3 * TAPF];
    const float r = w00 * v0 + w01 * v1 + w10 * v2 + w11 * v3;
    float mx = fmaxf(r, __shfl_xor(r, 1, 32));
    mx = fmaxf(mx, __shfl_xor(mx, 2, 32));
    if ((a & 3) == 0) outp[(size_t)it * OCS] = mx;   // lanes 0,4..28 -> 8 contiguous floats
  };

  // ---- software pipeline over bc, double-buffered via ASYNCcnt ----
  issue(0, 0);
  for (int it = 0; it < ITER - 1; ++it) {
    issue(it + 1, (uint32_t)(((it + 1) & 1) * (BUF * 4)));   // prefetch next bc
    // async loads complete in order: <=4 outstanding => batch 'it' is in LDS
    asm volatile("s_wait_asynccnt 0x4" ::: "memory");
    compute(it, (it & 1) * BUF);
  }
  asm volatile("s_wait_asynccnt 0x0" ::: "memory");
  compute(ITER - 1, ((ITER - 1) & 1) * BUF);
}

} // namespace

extern "C" void kernel_launch(void* const* d_in, const int* in_sizes, int n_in,
                              void* d_out, int out_size, void* d_ws, size_t ws_size,
                              hipStream_t stream) {
  (void)in_sizes; (void)n_in; (void)out_size; (void)d_ws; (void)ws_size;
  const float* in = (const float*)d_in[0];
  float* out = (float*)d_out;
  dim3 block(256);
  dim3 grid(H * W);          // 4096 blocks: one (y,x) each; 8 waves x 32 bc = 256 bc
  rotpool<<<grid, block, 0, stream>>>(in, out);
}